// Spe_RSA_42528766165344
// MI455X (gfx1250) — compile-verified
//
#include <hip/hip_runtime.h>
#include <math.h>

// ---- problem constants (match reference) ----
#define NB     4
#define DIMC   192
#define C3     576          // 3*DIM
#define HEADS  8
#define CPH    24           // DIM / HEADS
#define HEIGHT 256
#define WIDTH  256
#define HW     65536        // HEIGHT*WIDTH
#define GCHUNKS 16          // p-chunks per (b,h) for global gram reduction
#define GP      128         // p sub-tile for gram LDS staging
#define GSTR    129         // LDS row stride (conflict-free: 129 % 64 == 1)

typedef float v2f __attribute__((ext_vector_type(2)));
typedef float v8f __attribute__((ext_vector_type(8)));

// ---------------------------------------------------------------------------
// K1 / K6: 1x1 conv as GEMM with V_WMMA_F32_16X16X4_F32 (full fp32 precision).
// Y[b, m, p] = sum_k W[m,k] * X[b,k,p] + bias[m]
// One wave computes a 32(M) x 32(N) tile: 2x2 accumulators of 16x16.
// A frag (16x4 f32): lanes 0-15 rows M, v[0]=K+0,v[1]=K+1; lanes 16-31: K+2,K+3
// B frag (4x16 f32): lanes 0-15 cols N, v[0]=K+0 row, v[1]=K+1; lanes 16-31: K+2,K+3
// D (16x16 f32): vgpr r -> M=r (+8 for lanes>=16), N=lane&15
// ---------------------------------------------------------------------------
__global__ __launch_bounds__(256) void gemm1x1_wmma(
    const float* __restrict__ X, const float* __restrict__ Wm,
    const float* __restrict__ bias, float* __restrict__ Y,
    int M, int K)
{
  const int wave = threadIdx.x >> 5;
  const int lane = threadIdx.x & 31;
  const int l    = lane & 15;
  const int hi   = lane >> 4;                 // 0 or 1
  const int n0   = blockIdx.x * 256 + wave * 32;
  const int m0   = blockIdx.y * 32;
  const float* Xb = X + (size_t)blockIdx.z * (size_t)K * HW;
  float*       Yb = Y + (size_t)blockIdx.z * (size_t)M * HW;

  v8f acc00 = {}, acc01 = {}, acc10 = {}, acc11 = {};

  for (int k0 = 0; k0 < K; k0 += 4) {
    const int ka = k0 + (hi << 1);
    const float* wr0 = Wm + (size_t)(m0 + l) * K + ka;
    const float* wr1 = Wm + (size_t)(m0 + 16 + l) * K + ka;
    v2f a0, a1, b0, b1;
    a0[0] = wr0[0]; a0[1] = wr0[1];
    a1[0] = wr1[0]; a1[1] = wr1[1];
    const float* xc0 = Xb + (size_t)ka * HW + n0 + l;
    const float* xc1 = xc0 + 16;
    b0[0] = xc0[0]; b0[1] = xc0[HW];
    b1[0] = xc1[0]; b1[1] = xc1[HW];
    if (k0 + 8 < K) __builtin_prefetch(xc0 + 4 * HW, 0, 1);   // global_prefetch_b8

    acc00 = __builtin_amdgcn_wmma_f32_16x16x4_f32(false, a0, false, b0, (short)0, acc00, false, false);
    acc01 = __builtin_amdgcn_wmma_f32_16x16x4_f32(false, a0, false, b1, (short)0, acc01, false, false);
    acc10 = __builtin_amdgcn_wmma_f32_16x16x4_f32(false, a1, false, b0, (short)0, acc10, false, false);
    acc11 = __builtin_amdgcn_wmma_f32_16x16x4_f32(false, a1, false, b1, (short)0, acc11, false, false);
  }

  const int mofs = hi * 8;
  #pragma unroll
  for (int r = 0; r < 8; ++r) {
    const int row0 = m0 + r + mofs;
    const int row1 = m0 + 16 + r + mofs;
    const float bv0 = bias[row0], bv1 = bias[row1];
    Yb[(size_t)row0 * HW + n0 + l]      = acc00[r] + bv0;
    Yb[(size_t)row0 * HW + n0 + 16 + l] = acc01[r] + bv0;
    Yb[(size_t)row1 * HW + n0 + l]      = acc10[r] + bv1;
    Yb[(size_t)row1 * HW + n0 + 16 + l] = acc11[r] + bv1;
  }
}

// ---------------------------------------------------------------------------
// K2: depthwise 3x3 conv, SAME zero padding. One thread per output pixel.
// ---------------------------------------------------------------------------
__global__ __launch_bounds__(256) void dwconv3x3(
    const float* __restrict__ in, const float* __restrict__ w,
    const float* __restrict__ bias, float* __restrict__ out)
{
  const int x = threadIdx.x;    // 0..255 (full row)
  const int y = blockIdx.x;
  const int c = blockIdx.y;
  const int b = blockIdx.z;
  const float* ip = in + ((size_t)b * C3 + c) * HW;
  const float* wc = w + c * 9;
  float acc = bias[c];
  #pragma unroll
  for (int ky = 0; ky < 3; ++ky) {
    const int yy = y + ky - 1;
    if (yy < 0 || yy >= HEIGHT) continue;
    const float* row = ip + yy * WIDTH;
    #pragma unroll
    for (int kx = 0; kx < 3; ++kx) {
      const int xx = x + kx - 1;
      if (xx < 0 || xx >= WIDTH) continue;
      acc = fmaf(wc[ky * 3 + kx], row[xx], acc);
    }
  }
  out[((size_t)b * C3 + c) * HW + (size_t)y * WIDTH + x] = acc;
}

// ---------------------------------------------------------------------------
// K3: windowed channel attention. One 256-thread block per (window, head).
// q,k,v tiles are [24 x 64] (d = iy*8+ix within the 8x8 window).
// ---------------------------------------------------------------------------
__global__ __launch_bounds__(256) void win_attn(
    const float* __restrict__ qkv, const float* __restrict__ temp,
    float* __restrict__ o12)
{
  __shared__ float qs[CPH][65];     // pad 65: bank-conflict-free column reads
  __shared__ float ks[CPH][65];
  __shared__ float vs[CPH][65];
  __shared__ float sm[CPH][25];
  __shared__ float inq[CPH], ink[CPH];

  const int t  = threadIdx.x;
  const int wy = blockIdx.x >> 5;
  const int wx = blockIdx.x & 31;
  const int h  = blockIdx.y;
  const int b  = blockIdx.z;
  const size_t base = (size_t)b * C3 * HW;

  for (int i = t; i < CPH * 64; i += 256) {
    const int c = i >> 6, d = i & 63;
    const int iy = d >> 3, ix = d & 7;
    const size_t pix = (size_t)(wy * 8 + iy) * WIDTH + (wx * 8 + ix);
    qs[c][d] = qkv[base + (size_t)(h * CPH + c) * HW + pix];
    ks[c][d] = qkv[base + (size_t)(DIMC + h * CPH + c) * HW + pix];
    vs[c][d] = qkv[base + (size_t)(2 * DIMC + h * CPH + c) * HW + pix];
  }
  __syncthreads();

  if (t < 48) {                       // L2 row norms of q and k
    const int r = t % 24;
    float s = 0.f;
    if (t < 24) { for (int d = 0; d < 64; ++d) s = fmaf(qs[r][d], qs[r][d], s); }
    else        { for (int d = 0; d < 64; ++d) s = fmaf(ks[r][d], ks[r][d], s); }
    const float n = fmaxf(sqrtf(s), 1e-12f);
    if (t < 24) inq[r] = 1.f / n; else ink[r] = 1.f / n;
  }
  __syncthreads();

  const float tmp = temp[h];
  for (int i = t; i < CPH * CPH; i += 256) {   // logits (576 pairs)
    const int c = i / 24, e = i % 24;
    float s = 0.f;
    for (int d = 0; d < 64; ++d) s = fmaf(qs[c][d], ks[e][d], s);
    sm[c][e] = s * inq[c] * ink[e] * tmp;
  }
  __syncthreads();

  if (t < 24) {                       // stable softmax over e
    float mx = -3.4e38f;
    for (int e = 0; e < 24; ++e) mx = fmaxf(mx, sm[t][e]);
    float ssum = 0.f;
    for (int e = 0; e < 24; ++e) { const float ev = expf(sm[t][e] - mx); sm[t][e] = ev; ssum += ev; }
    const float inv = 1.f / ssum;
    for (int e = 0; e < 24; ++e) sm[t][e] *= inv;
  }
  __syncthreads();

  const size_t obase = (size_t)b * DIMC * HW;
  for (int i = t; i < CPH * 64; i += 256) {    // o1 = attn @ v
    const int c = i >> 6, d = i & 63;
    float s = 0.f;
    for (int e = 0; e < 24; ++e) s = fmaf(sm[c][e], vs[e][d], s);
    const int iy = d >> 3, ix = d & 7;
    const size_t pix = (size_t)(wy * 8 + iy) * WIDTH + (wx * 8 + ix);
    o12[obase + (size_t)(h * CPH + c) * HW + pix] = s;
  }
}

// ---------------------------------------------------------------------------
// K4a: global-branch Gram partials. One block per (b, h, p-chunk of 4096).
// Accumulates S[c,e] (576), sum q^2 (24), sum k^2 (24) -> 624 values/chunk.
// ---------------------------------------------------------------------------
__global__ __launch_bounds__(256) void gram_partial(
    const float* __restrict__ qkv, float* __restrict__ part)
{
  __shared__ float qs[CPH][GSTR];
  __shared__ float ks[CPH][GSTR];
  const int t = threadIdx.x;
  const int chunk = blockIdx.x;
  const int h = blockIdx.y, b = blockIdx.z;
  const size_t qbase = (size_t)b * C3 * HW + (size_t)(h * CPH) * HW;
  const size_t kbase = qbase + (size_t)DIMC * HW;
  const int p0 = chunk * (HW / GCHUNKS);
  const int row2 = t >> 7;      // 0/1
  const int col  = t & 127;

  float a0 = 0.f, a1 = 0.f, a2 = 0.f;

  for (int sub = 0; sub < (HW / GCHUNKS) / GP; ++sub) {
    const int p = p0 + sub * GP;
    __syncthreads();
    for (int r = row2; r < CPH; r += 2) {
      qs[r][col] = qkv[qbase + (size_t)r * HW + p + col];
      ks[r][col] = qkv[kbase + (size_t)r * HW + p + col];
    }
    __syncthreads();

    {   // idx = t  (always a (c,e) pair)
      const int c = t / 24, e = t % 24;
      float s = 0.f;
      for (int d = 0; d < GP; ++d) s = fmaf(qs[c][d], ks[e][d], s);
      a0 += s;
    }
    {   // idx = t + 256 (always a pair: 256..511 < 576)
      const int idx = t + 256;
      const int c = idx / 24, e = idx % 24;
      float s = 0.f;
      for (int d = 0; d < GP; ++d) s = fmaf(qs[c][d], ks[e][d], s);
      a1 += s;
    }
    {   // idx = t + 512: pairs for t<64, norms for 64<=t<112
      const int idx = t + 512;
      if (idx < 576) {
        const int c = idx / 24, e = idx % 24;
        float s = 0.f;
        for (int d = 0; d < GP; ++d) s = fmaf(qs[c][d], ks[e][d], s);
        a2 += s;
      } else if (idx < 600) {
        const int r = idx - 576;
        float s = 0.f;
        for (int d = 0; d < GP; ++d) s = fmaf(qs[r][d], qs[r][d], s);
        a2 += s;
      } else if (idx < 624) {
        const int r = idx - 600;
        float s = 0.f;
        for (int d = 0; d < GP; ++d) s = fmaf(ks[r][d], ks[r][d], s);
        a2 += s;
      }
    }
  }

  const size_t pbase = ((size_t)(b * HEADS + h) * GCHUNKS + chunk) * 640;
  part[pbase + t]       = a0;
  part[pbase + t + 256] = a1;
  if (t < 112) part[pbase + t + 512] = a2;
}

// ---------------------------------------------------------------------------
// K4b: reduce partials, normalize, softmax -> attn[b,h,24,24]
// ---------------------------------------------------------------------------
__global__ __launch_bounds__(256) void gram_finalize(
    const float* __restrict__ part, const float* __restrict__ temp,
    float* __restrict__ attn)
{
  __shared__ float S[CPH][25];
  __shared__ float nq[CPH], nk[CPH];
  const int t = threadIdx.x;
  const int b = blockIdx.x >> 3, h = blockIdx.x & 7;
  const size_t pbase = (size_t)(b * HEADS + h) * GCHUNKS * 640;

  for (int idx = t; idx < 624; idx += 256) {
    float s = 0.f;
    for (int ch = 0; ch < GCHUNKS; ++ch) s += part[pbase + (size_t)ch * 640 + idx];
    if (idx < 576)      S[idx / 24][idx % 24] = s;
    else if (idx < 600) nq[idx - 576] = fmaxf(sqrtf(s), 1e-12f);
    else                nk[idx - 600] = fmaxf(sqrtf(s), 1e-12f);
  }
  __syncthreads();

  if (t < 24) {
    const float tmp = temp[h];
    float row[24];
    float mx = -3.4e38f;
    for (int e = 0; e < 24; ++e) {
      const float v = S[t][e] / (nq[t] * nk[e]) * tmp;
      row[e] = v; mx = fmaxf(mx, v);
    }
    float ssum = 0.f;
    for (int e = 0; e < 24; ++e) { row[e] = expf(row[e] - mx); ssum += row[e]; }
    const float inv = 1.f / ssum;
    float* ap = attn + (size_t)(b * HEADS + h) * 576 + t * 24;
    for (int e = 0; e < 24; ++e) ap[e] = row[e] * inv;
  }
}

// ---------------------------------------------------------------------------
// K5: apply global attn per pixel and add into o1 (in place): o12 += attn @ v
// One block per (b, h, 256-pixel tile).
// ---------------------------------------------------------------------------
__global__ __launch_bounds__(256) void global_apply(
    const float* __restrict__ qkv, const float* __restrict__ attn,
    float* __restrict__ o12)
{
  __shared__ float vs[CPH][256];
  __shared__ float am[CPH][25];
  const int t = threadIdx.x;
  const int p0 = blockIdx.x * 256;
  const int h = blockIdx.y, b = blockIdx.z;
  const size_t vbase = (size_t)b * C3 * HW + (size_t)(2 * DIMC + h * CPH) * HW;

  for (int r = 0; r < CPH; ++r) vs[r][t] = qkv[vbase + (size_t)r * HW + p0 + t];
  for (int i = t; i < 576; i += 256) am[i / 24][i % 24] = attn[(size_t)(b * HEADS + h) * 576 + i];
  __syncthreads();

  const size_t obase = (size_t)b * DIMC * HW + (size_t)(h * CPH) * HW + p0 + t;
  #pragma unroll
  for (int c = 0; c < CPH; ++c) {
    float s = 0.f;
    for (int e = 0; e < CPH; ++e) s = fmaf(am[c][e], vs[e][t], s);
    o12[obase + (size_t)c * HW] += s;
  }
}

// ---------------------------------------------------------------------------
extern "C" void kernel_launch(void* const* d_in, const int* in_sizes, int n_in,
                              void* d_out, int out_size, void* d_ws, size_t ws_size,
                              hipStream_t stream) {
  (void)in_sizes; (void)n_in; (void)out_size; (void)ws_size;
  const float* x      = (const float*)d_in[0];
  const float* temp   = (const float*)d_in[1];
  const float* qkv_w  = (const float*)d_in[2];   // [576,192]
  const float* qkv_b  = (const float*)d_in[3];
  const float* dw_w   = (const float*)d_in[4];   // [576,3,3]
  const float* dw_b   = (const float*)d_in[5];
  const float* proj_w = (const float*)d_in[6];   // [192,192]
  const float* proj_b = (const float*)d_in[7];
  float* out = (float*)d_out;
  float* ws  = (float*)d_ws;

  const size_t QKV_ELEMS = (size_t)NB * C3 * HW;         // 150,994,944 floats
  float* qkv_pre  = ws;                                  // 604 MB
  float* qkv_post = ws + QKV_ELEMS;                      // 604 MB
  float* o12      = ws;                                  // reuse qkv_pre region
  float* gpart    = ws + 2 * QKV_ELEMS;                  // 32*16*640 floats
  float* attn     = gpart + (size_t)NB * HEADS * GCHUNKS * 640;

  // 1) qkv = 1x1 conv (WMMA f32 GEMM, fused bias)
  gemm1x1_wmma<<<dim3(HW / 256, C3 / 32, NB), 256, 0, stream>>>(
      x, qkv_w, qkv_b, qkv_pre, C3, DIMC);
  // 2) depthwise 3x3, SAME
  dwconv3x3<<<dim3(HEIGHT, C3, NB), 256, 0, stream>>>(
      qkv_pre, dw_w, dw_b, qkv_post);
  // 3) windowed branch -> o1 (writes every pixel once)
  win_attn<<<dim3(1024, HEADS, NB), 256, 0, stream>>>(qkv_post, temp, o12);
  // 4) global branch: Gram partials -> attn matrices
  gram_partial<<<dim3(GCHUNKS, HEADS, NB), 256, 0, stream>>>(qkv_post, gpart);
  gram_finalize<<<dim3(NB * HEADS), 256, 0, stream>>>(gpart, temp, attn);
  // 5) o12 = o1 + o2
  global_apply<<<dim3(HW / 256, HEADS, NB), 256, 0, stream>>>(qkv_post, attn, o12);
  // 6) proj 1x1 conv -> out
  gemm1x1_wmma<<<dim3(HW / 256, DIMC / 32, NB), 256, 0, stream>>>(
      o12, proj_w, proj_b, out, DIMC, DIMC);
}